// MeshNN_2D_26010321944828
// MI455X (gfx1250) — compile-verified
//
#include <hip/hip_runtime.h>
#include <hip/hip_bf16.h>

// FEM barycentric interpolation at triangle centroids.
// inputs (setup_inputs order):
//   d_in[0] coordinates : float32 (NX*NY, 2)
//   d_in[1] nodal_values: float32 (2, NX*NY)
//   d_in[2] connectivity: int32   (n_elem, 3), 1-based node ids
//   d_in[3] cell_id     : int32   (n_elem,)
// output: float32 (2, n_elem) flattened.

#define PF_DIST 3072   // elements of lookahead on the pure streams

__global__ __launch_bounds__(256)
void MeshNN_2D_kernel(const float* __restrict__ coords,   // (N,2)
                      const float* __restrict__ nodal,    // (2,N)
                      const int*   __restrict__ conn,     // (E,3) 1-based
                      const int*   __restrict__ cell_id,  // (E,)
                      float*       __restrict__ out,      // (2,E)
                      int n_elem, int n_nodes)
{
    const int e = blockIdx.x * blockDim.x + threadIdx.x;
    if (e >= n_elem) return;

    // Prefetch the two streamed-once inputs far ahead (-> global_prefetch_b8).
    // Speculative prefetch is silently dropped on invalid translation, so the
    // unguarded tail is safe per the CDNA5 prefetch semantics.
    __builtin_prefetch(cell_id + e + PF_DIST, 0, 0);
    __builtin_prefetch(conn + 3 * (e + PF_DIST), 0, 0);

    // Streamed-once data: non-temporal loads keep L2/WGP$ for the reused
    // coordinate/nodal arrays.
    const int  c  = __builtin_nontemporal_load(cell_id + e);
    const long cb = 3L * (long)c;
    const int n0 = __builtin_nontemporal_load(conn + cb + 0) - 1;
    const int n1 = __builtin_nontemporal_load(conn + cb + 1) - 1;
    const int n2 = __builtin_nontemporal_load(conn + cb + 2) - 1;

    // Reused data (each node hit by ~6 triangles): regular temporal loads.
    const float2* __restrict__ cxy = (const float2*)coords;
    const float2 p1 = cxy[n0];
    const float2 p2 = cxy[n1];
    const float2 p3 = cxy[n2];

    const float x1 = p1.x, y1 = p1.y;
    const float x2 = p2.x, y2 = p2.y;
    const float x3 = p3.x, y3 = p3.y;

    const float third = 1.0f / 3.0f;
    const float xg = third * (x1 + x2 + x3);
    const float yg = third * (y1 + y2 + y3);

    // Barycentric coordinates of the centroid (analytic inverse of the
    // affine map; single shared divisor d as in the reference).
    const float d  = x1 * (y3 - y2) + x2 * (y1 - y3) + x3 * (y2 - y1);
    const float rd = 1.0f / d;

    const float N0 = (xg * (y3 - y2) + yg * (x2 - x3) + (x3 * y2 - x2 * y3)) * rd;
    const float N1 = (xg * (y1 - y3) + yg * (x3 - x1) + (x1 * y3 - x3 * y1)) * rd;
    const float N2 = (xg * (y2 - y1) + yg * (x1 - x2) + (x2 * y1 - x1 * y2)) * rd;

    // Two-component interpolation (gathers from the reused nodal array).
    const float v10 = nodal[n0];
    const float v11 = nodal[n1];
    const float v12 = nodal[n2];
    const float v20 = nodal[n_nodes + n0];
    const float v21 = nodal[n_nodes + n1];
    const float v22 = nodal[n_nodes + n2];

    const float u0 = N0 * v10 + N1 * v11 + N2 * v12;
    const float u1 = N0 * v20 + N1 * v21 + N2 * v22;

    // Output is written once and never read back: NT stores (TH=NT) so the
    // write stream does not evict the gather working set from L2.
    __builtin_nontemporal_store(u0, out + e);
    __builtin_nontemporal_store(u1, out + n_elem + e);
}

extern "C" void kernel_launch(void* const* d_in, const int* in_sizes, int n_in,
                              void* d_out, int out_size, void* d_ws, size_t ws_size,
                              hipStream_t stream) {
    const float* coords  = (const float*)d_in[0];
    const float* nodal   = (const float*)d_in[1];
    const int*   conn    = (const int*)d_in[2];
    const int*   cell_id = (const int*)d_in[3];
    float* out = (float*)d_out;

    const int n_elem  = in_sizes[3];      // length of cell_id
    const int n_nodes = in_sizes[0] / 2;  // coordinates is (N,2)

    const int block = 256;                 // 8 wave32s
    const int grid  = (n_elem + block - 1) / block;
    MeshNN_2D_kernel<<<grid, block, 0, stream>>>(coords, nodal, conn, cell_id,
                                                 out, n_elem, n_nodes);
}